// STAddAttention_85899345920234
// MI455X (gfx1250) — compile-verified
//
#include <hip/hip_runtime.h>

typedef __attribute__((ext_vector_type(2))) float v2f;
typedef __attribute__((ext_vector_type(8))) float v8f;

#define B_   4
#define N_   207
#define BN   828      // B_*N_
#define L_   96
#define S_   96
#define H_   8
#define E_   64
#define D_   64
#define HE   512      // H_*E_ : row stride (floats) in q/k/v
#define LBLK 32       // L rows per workgroup
#define ROWP 100      // padded LDS row stride in floats (breaks 64-bank conflicts)

// Fused STAddAttention:
//   scores = LayerNorm_H( QK^T + mask + attn_scores )   -> out[0]  [B,N,H,L,S]
//   V      = softmax_s(0.125*scores) @ values           -> out[1]  [B,N,L,H,D]
// One workgroup = one (b*n, 32-row L block); wave w == head w.
__global__ __launch_bounds__(256)
void STAddAttention_fused_kernel(const float* __restrict__ q,
                                 const float* __restrict__ k,
                                 const float* __restrict__ v,
                                 const float* __restrict__ mask,
                                 const float* __restrict__ ascores,
                                 const float* __restrict__ gamma,
                                 const float* __restrict__ beta,
                                 float* __restrict__ out_scores,
                                 float* __restrict__ out_v)
{
    extern __shared__ float lds[];          // [H_][LBLK][ROWP] = 100 KB
    const int tid  = threadIdx.x;
    const int lane = tid & 31;
    const int h    = tid >> 5;              // wave id == head id
    const int n16  = lane & 15;
    const int hv   = lane >> 4;             // 0: lanes 0-15, 1: lanes 16-31
    const int l0   = blockIdx.x * LBLK;
    const int bn   = blockIdx.y;

#define SC(hh,row,col) lds[(((hh)*LBLK + (row))*ROWP) + (col)]

    const float* qbase = q + ((size_t)bn * L_ + l0) * HE + h * E_;
    const float* kbase = k + (size_t)bn * S_ * HE + h * E_;

    // ---------------- Phase 1: x = Q K^T + mask + attn_scores  ->  LDS -------
    // WMMA f32 16x16x4 fragment layout (ISA 7.12.2):
    //   A (16x4): lane m(<16) holds A[m,k0],A[m,k0+1]; lane m+16 holds A[m,k0+2],A[m,k0+3]
    //   B (4x16): lane n(<16) holds B[k0,n],B[k0+1,n]; lane n+16 holds B[k0+2,n],B[k0+3,n]
    //   C/D: vgpr r -> row r (lanes 0-15) / row r+8 (lanes 16-31), col = lane&15
    for (int mi = 0; mi < 2; ++mi) {
        const float* qr = qbase + (size_t)(mi * 16 + n16) * HE;
        v2f aq[16];
#pragma unroll
        for (int kk = 0; kk < 16; ++kk)
            aq[kk] = *(const v2f*)(qr + kk * 4 + hv * 2);

        for (int st = 0; st < 6; ++st) {
            const int col = st * 16 + n16;
            v8f c;
#pragma unroll
            for (int r = 0; r < 8; ++r) {   // C init = additive terms (fused)
                const int row = mi * 16 + hv * 8 + r;
                c[r] = mask[(l0 + row) * S_ + col] +
                       ascores[(((size_t)bn * H_ + h) * L_ + (l0 + row)) * S_ + col];
            }
            const float* kr = kbase + (size_t)(st * 16 + n16) * HE;
#pragma unroll
            for (int kk = 0; kk < 16; ++kk) {
                v2f b = *(const v2f*)(kr + kk * 4 + hv * 2);
                c = __builtin_amdgcn_wmma_f32_16x16x4_f32(false, aq[kk], false, b,
                                                          (short)0, c, false, false);
            }
#pragma unroll
            for (int r = 0; r < 8; ++r) {
                const int row = mi * 16 + hv * 8 + r;
                SC(h, row, col) = c[r];
            }
        }
    }
    __syncthreads();

    // ---------------- Phase 2: LayerNorm across the 8 heads ------------------
    float g[H_], bt[H_];
#pragma unroll
    for (int hh = 0; hh < H_; ++hh) { g[hh] = gamma[hh]; bt[hh] = beta[hh]; }

    for (int i = 0; i < (LBLK * S_) / 256; ++i) {     // 12 positions per thread
        const int pos = tid + i * 256;
        const int l = pos / S_;
        const int s = pos - l * S_;
        float xv[H_];
        float sum = 0.f, sq = 0.f;
#pragma unroll
        for (int hh = 0; hh < H_; ++hh) {
            xv[hh] = SC(hh, l, s);
            sum += xv[hh];
            sq  += xv[hh] * xv[hh];
        }
        const float mean = sum * (1.0f / H_);
        const float var  = sq * (1.0f / H_) - mean * mean;
        const float rs   = rsqrtf(var + 1e-5f);
#pragma unroll
        for (int hh = 0; hh < H_; ++hh) {
            const float y = (xv[hh] - mean) * rs * g[hh] + bt[hh];
            SC(hh, l, s) = y;
            out_scores[(((size_t)bn * H_ + hh) * L_ + (l0 + l)) * S_ + s] = y;
        }
    }
    __syncthreads();

    // ---------------- Phase 3: softmax over s, row (h, l=lane) ---------------
    {
        float* row = &SC(h, lane, 0);
        const float temp = 0.125f;                    // 1/sqrt(E)
        float mx = -3.402823466e+38f;
        for (int s = 0; s < S_; ++s) mx = fmaxf(mx, row[s] * temp);
        float sum = 0.f;
        for (int s = 0; s < S_; ++s) {
            const float e = __expf(row[s] * temp - mx);
            row[s] = e;
            sum += e;
        }
        const float inv = 1.0f / sum;
        for (int s = 0; s < S_; ++s) row[s] *= inv;
    }
    __syncthreads();

    // ---------------- Phase 4: O = P @ V  -> out_v [B,N,L,H,D] ---------------
    const float* vbase = v + (size_t)bn * S_ * HE + h * D_;
    for (int mi = 0; mi < 2; ++mi) {
        v2f ap[24];
#pragma unroll
        for (int kk = 0; kk < 24; ++kk)               // A frags from padded LDS
            ap[kk] = *(const v2f*)&SC(h, mi * 16 + n16, kk * 4 + hv * 2);

        for (int nt = 0; nt < 4; ++nt) {
            v8f c = {};
            const float* vb = vbase + nt * 16 + n16;
#pragma unroll
            for (int kk = 0; kk < 24; ++kk) {
                const int s0 = kk * 4 + hv * 2;
                v2f b;
                b.x = vb[(size_t)s0 * HE];
                b.y = vb[(size_t)(s0 + 1) * HE];
                c = __builtin_amdgcn_wmma_f32_16x16x4_f32(false, ap[kk], false, b,
                                                          (short)0, c, false, false);
            }
            const int d = nt * 16 + n16;
#pragma unroll
            for (int r = 0; r < 8; ++r) {
                const int row = mi * 16 + hv * 8 + r;
                out_v[(((size_t)bn * L_ + (l0 + row)) * H_ + h) * D_ + d] = c[r];
            }
        }
    }
#undef SC
}

extern "C" void kernel_launch(void* const* d_in, const int* in_sizes, int n_in,
                              void* d_out, int out_size, void* d_ws, size_t ws_size,
                              hipStream_t stream) {
    (void)in_sizes; (void)n_in; (void)d_ws; (void)ws_size; (void)out_size;
    const float* q  = (const float*)d_in[0];
    const float* k  = (const float*)d_in[1];
    const float* v  = (const float*)d_in[2];
    const float* mk = (const float*)d_in[3];
    const float* as = (const float*)d_in[4];
    const float* gm = (const float*)d_in[5];
    const float* bt = (const float*)d_in[6];
    float* out_scores = (float*)d_out;                        // [B,N,H,L,S]
    float* out_v = out_scores + (size_t)BN * H_ * L_ * S_;    // [B,N,L,H,D]

    dim3 grid(L_ / LBLK, BN);                                 // 3 x 828
    const size_t shmem = (size_t)H_ * LBLK * ROWP * sizeof(float);  // 100 KB
    STAddAttention_fused_kernel<<<grid, 256, shmem, stream>>>(
        q, k, v, mk, as, gm, bt, out_scores, out_v);
}